// NearestInterpolator_Torch_28011776704992
// MI455X (gfx1250) — compile-verified
//
#include <hip/hip_runtime.h>

typedef __attribute__((ext_vector_type(2))) float v2f;
typedef __attribute__((ext_vector_type(8))) float v8f;

#define M_POINTS  8192
#define N_QUERIES 16384
#define BLOCK_THREADS 256
#define WAVES_PER_BLOCK (BLOCK_THREADS / 32)

// 1-NN via WMMA: per wave, 16 queries vs all 8192 points.
// A (16x4 f32) = -2 * Q tile (K=2,3 zeroed by zeroing lanes 16-31 of A),
// B (4x16 f32) = P tile from LDS, D = -2*Q.P^T per 16x16 tile.
// Candidate = ||p||^2 + D ; running (min, idx) kept per C/D slot, reduced
// across the 16 column lanes at the end with shfl_xor butterflies.
__launch_bounds__(BLOCK_THREADS, 2)
__global__ void nn1_wmma_kernel(const float* __restrict__ points_q,
                                const float* __restrict__ points,
                                const float* __restrict__ values,
                                float* __restrict__ out) {
    __shared__ float2 s_xy[M_POINTS];   // 64 KB of the 320 KB WGP LDS

    const int tid = threadIdx.x;

    // Cooperative stage of all points into LDS (each thread: 32 points).
    const float2* p2 = (const float2*)points;
    for (int i = tid; i < M_POINTS; i += BLOCK_THREADS) {
        s_xy[i] = p2[i];
    }
    __syncthreads();

    const int lane  = tid & 31;
    const int wave  = tid >> 5;
    const int col   = lane & 15;              // column of the 16x16 tile
    const int half  = lane >> 4;              // 0: rows 0-7, 1: rows 8-15
    const int qbase = (blockIdx.x * WAVES_PER_BLOCK + wave) * 16;

    // A matrix: lanes 0-15 hold (K=0,K=1) = (-2qx, -2qy); lanes 16-31 hold
    // (K=2,K=3) = 0 so B's upper K rows are don't-care.
    float2 q = ((const float2*)points_q)[qbase + col];
    v2f a;
    a.x = (lane < 16) ? (-2.0f * q.x) : 0.0f;
    a.y = (lane < 16) ? (-2.0f * q.y) : 0.0f;

    float best[8];
    int   bidx[8];
#pragma unroll
    for (int r = 0; r < 8; ++r) { best[r] = 3.402823466e38f; bidx[r] = 0; }

#pragma unroll 4
    for (int t = 0; t < M_POINTS; t += 16) {
        const int pi = t + col;               // lanes 16-31 duplicate, harmless
        float2 p = s_xy[pi];                  // ds_load_b64
        float  pn = __builtin_fmaf(p.y, p.y, p.x * p.x);

        v2f b;
        b.x = p.x;                            // K=0 (lanes 0-15) / K=2 (16-31)
        b.y = p.y;                            // K=1 (lanes 0-15) / K=3 (16-31)

        v8f c = {};
        c = __builtin_amdgcn_wmma_f32_16x16x4_f32(
                /*neg_a=*/false, a, /*neg_b=*/false, b,
                /*c_mod=*/(short)0, c, /*reuse_a=*/false, /*reuse_b=*/false);

#pragma unroll
        for (int r = 0; r < 8; ++r) {
            float cand = pn + c[r];           // ||p||^2 - 2 q.p
            bool lt = cand < best[r];         // strict <: keeps earliest index
            best[r] = lt ? cand : best[r];
            bidx[r] = lt ? pi   : bidx[r];
        }
    }

    // Cross-lane argmin over the 16 lanes sharing each row (masks <= 8 stay
    // inside each 16-lane group under wave32).
#pragma unroll
    for (int m = 8; m >= 1; m >>= 1) {
#pragma unroll
        for (int r = 0; r < 8; ++r) {
            float ob = __shfl_xor(best[r], m, 32);
            int   oi = __shfl_xor(bidx[r], m, 32);
            if (ob < best[r] || (ob == best[r] && oi < bidx[r])) {
                best[r] = ob;
                bidx[r] = oi;
            }
        }
    }

    // Lane 0 holds rows 0-7, lane 16 holds rows 8-15.
    if (col == 0) {
#pragma unroll
        for (int r = 0; r < 8; ++r) {
            out[qbase + r + 8 * half] = values[bidx[r]];
        }
    }
}

extern "C" void kernel_launch(void* const* d_in, const int* in_sizes, int n_in,
                              void* d_out, int out_size, void* d_ws, size_t ws_size,
                              hipStream_t stream) {
    const float* points_q = (const float*)d_in[0];  // [16384, 2] f32
    const float* points   = (const float*)d_in[1];  // [8192, 2]  f32
    const float* values   = (const float*)d_in[2];  // [8192]     f32
    float* out            = (float*)d_out;          // [16384]    f32

    const int queries_per_block = WAVES_PER_BLOCK * 16;       // 128
    const int grid = N_QUERIES / queries_per_block;           // 128 blocks

    nn1_wmma_kernel<<<grid, BLOCK_THREADS, 0, stream>>>(points_q, points, values, out);
}